// AutoregressiveBisectionInverter_50328426774817
// MI455X (gfx1250) — compile-verified
//
#include <hip/hip_runtime.h>
#include <math.h>

// -------- CDNA5 WMMA types --------
typedef __attribute__((ext_vector_type(2))) float v2f;
typedef __attribute__((ext_vector_type(8))) float v8f;

#define DD    32
#define HALF  16
#define TOLF  1e-6f

// sign() -> {-1,0,1} as int (matches jnp.sign on floats) — adaptation loop only
__device__ __forceinline__ int isgn(float v) { return (v > 0.f) - (v < 0.f); }

// jax.nn.softplus(x) = max(x,0) + log1p(exp(-|x|))  (stable both ways)
__device__ __forceinline__ float softplus_f(float v) {
    return fmaxf(v, 0.f) + log1pf(expf(-fabsf(v)));
}

// Hardware tanh (V_TANH_F32 is a gfx1250 TRANS op) if the builtin exists;
// otherwise fall back to libm's branchy expansion.
__device__ __forceinline__ float fast_tanh(float v) {
#if __has_builtin(__builtin_amdgcn_tanhf)
    return __builtin_amdgcn_tanhf(v);
#else
    return tanhf(v);
#endif
}

// g(x) = a*x + 0.1*x^3 + tmy   (monotone increasing; tmy = tanh(c+b_i) - y_i)
__device__ __forceinline__ float g_eval(float a, float tmy, float x) {
    return fmaf(0.1f * x, x * x, fmaf(a, x, tmy));
}

// Wave32 lane-XOR-16 via ds_swizzle SWAPX16 (and=0x1f, or=0, xor=0x10).
__device__ __forceinline__ float swap16(float v) {
    return __int_as_float(__builtin_amdgcn_ds_swizzle(__float_as_int(v), 0x401f));
}

// Faithful port of _adapt_interval + _bisection_search for this scalar family.
__device__ float solve_root(float a, float tmy) {
    float lo = -10.f, up = 10.f;
    int ls = isgn(g_eval(a, tmy, lo));
    int us = isgn(g_eval(a, tmy, up));
    float ex = up - lo;
    while (ls == us) {                       // expand until signs differ
        float lo_new = (ls == 1) ? (lo - ex) : up;
        float up_new = (ls == 1) ? lo : (up + ex);
        lo = lo_new; up = up_new;
        ex *= 2.f;
        ls = isgn(g_eval(a, tmy, lo));
        us = isgn(g_eval(a, tmy, up));
    }
    if (us == 0) lo = up;
    if (ls == 0) up = lo;
    int it = 0;
    while ((up - lo > 2.f * TOLF) && (it < 100)) {
        const float mid = 0.5f * (lo + up);
        const float gm  = g_eval(a, tmy, mid);
        const bool  pos = gm > 0.f;          // sign == 1
        const bool  zer = gm == 0.f;         // sign == 0
        const float nlo = pos ? lo : mid;
        const float nup = pos ? mid : up;
        lo = zer ? mid : nlo;
        up = zer ? mid : nup;
        ++it;
    }
    return 0.5f * (lo + up);
}

// One wave (32 lanes) inverts 16 batch rows. Lanes 16..31 mirror lanes 0..15
// (same batch) so that the f32 WMMA A/B operands, whose K dimension is split
// across wave halves, are pure per-lane register selects.
__global__ __launch_bounds__(32)
void bnaf_invert_kernel(const float* __restrict__ y,
                        const float* __restrict__ W,
                        const float* __restrict__ s,
                        const float* __restrict__ bvec,
                        float* __restrict__ out,
                        int nbatch) {
    const int lane  = threadIdx.x;       // 0..31
    const int m     = lane & 15;         // batch slot within tile
    const int hi    = lane >> 4;         // wave half (K-offset for WMMA)
    int batch       = blockIdx.x * HALF + m;
    const bool live = (batch < nbatch);
    if (batch >= nbatch) batch = nbatch - 1;   // clamp loads, keep EXEC full

    // ---- preload this batch's y row (static-register array) ----
    float yv[DD];
#pragma unroll
    for (int j = 0; j < DD; j += 4) {
        const float4 t = *(const float4*)(y + batch * DD + j);
        yv[j] = t.x; yv[j + 1] = t.y; yv[j + 2] = t.z; yv[j + 3] = t.w;
    }

    // ---- hoist all softplus(s_i) off the serial chain (independent of x) ----
    float aarr[DD];
#pragma unroll
    for (int i = 0; i < DD; ++i) aarr[i] = softplus_f(s[i]);

    float xv[DD];
    float cpart[HALF];
#pragma unroll
    for (int k = 0; k < HALF; ++k) cpart[k] = 0.f;

    // ================= Phase 1: rows 0..15 (serial, rank-1 updates) =========
#pragma unroll
    for (int i = 0; i < HALF; ++i) {
        const float t = fast_tanh(cpart[i] + bvec[i]);
        const float r = solve_root(aarr[i], t - yv[i]);
        xv[i] = r;
#pragma unroll
        for (int k = i + 1; k < HALF; ++k)
            cpart[k] = fmaf(W[k * DD + i], r, cpart[k]);
    }

    // ============ Phase 2: cross-block coupling via f32 WMMA ================
    // Ccross[i2][b] = sum_{j<16} W[16+i2][j] * x[b][j]
    //   A (16x4, M=i2, K=j): lane -> W[16+m][k0 + 2*hi .. +1]   (b64 load)
    //   B (4x16, K=j, N=b) : lane -> x_m[k0 + 2*hi .. +1]       (reg select)
    v8f acc = {};
#pragma unroll
    for (int kc = 0; kc < 4; ++kc) {
        const int k0 = kc * 4;
        const float2 wp = *(const float2*)(W + (HALF + m) * DD + k0 + 2 * hi);
        v2f av; av.x = wp.x; av.y = wp.y;
        v2f bv; bv.x = hi ? xv[k0 + 2] : xv[k0 + 0];
                bv.y = hi ? xv[k0 + 3] : xv[k0 + 1];
        acc = __builtin_amdgcn_wmma_f32_16x16x4_f32(
            /*neg_a=*/false, av, /*neg_b=*/false, bv,
            /*c_mod=*/(short)0, acc, /*reuse_a=*/false, /*reuse_b=*/false);
    }

    // D layout: VGPR r, lanes 0-15 -> (M=r, N=lane); lanes 16-31 -> (M=r+8).
    // With M=i2, N=batch: lane b holds i2=0..7 low half / i2=8..15 high half.
    // One ds_swizzle SWAPX16 per register gives every lane all 16 cross terms.
    float cross[HALF];
#pragma unroll
    for (int r = 0; r < 8; ++r) {
        const float mine  = acc[r];
        const float other = swap16(mine);
        cross[r]     = hi ? other : mine;
        cross[r + 8] = hi ? mine  : other;
    }

    // ================= Phase 3: rows 16..31 =================================
    float cpart2[HALF];
#pragma unroll
    for (int k = 0; k < HALF; ++k) cpart2[k] = 0.f;
#pragma unroll
    for (int i2 = 0; i2 < HALF; ++i2) {
        const int   i = HALF + i2;
        const float c = cross[i2] + cpart2[i2];
        const float t = fast_tanh(c + bvec[i]);
        const float r = solve_root(aarr[i], t - yv[i]);
        xv[i] = r;
#pragma unroll
        for (int k = i2 + 1; k < HALF; ++k)
            cpart2[k] = fmaf(W[(HALF + k) * DD + i], r, cpart2[k]);
    }

    // ---- store (low half only; high half is an exact mirror) ----
    if (live && hi == 0) {
#pragma unroll
        for (int j = 0; j < DD; j += 4) {
            float4 t; t.x = xv[j]; t.y = xv[j + 1]; t.z = xv[j + 2]; t.w = xv[j + 3];
            *(float4*)(out + batch * DD + j) = t;
        }
    }
}

extern "C" void kernel_launch(void* const* d_in, const int* in_sizes, int n_in,
                              void* d_out, int out_size, void* d_ws, size_t ws_size,
                              hipStream_t stream) {
    const float* y = (const float*)d_in[0];   // (B, 32)
    const float* W = (const float*)d_in[1];   // (32, 32) strictly lower-tri
    const float* s = (const float*)d_in[2];   // (32,)
    const float* b = (const float*)d_in[3];   // (32,)
    float* out     = (float*)d_out;           // (B, 32)

    const int B = in_sizes[0] / DD;           // 1024
    const int tiles = (B + HALF - 1) / HALF;  // 64 single-wave workgroups
    bnaf_invert_kernel<<<dim3(tiles), dim3(32), 0, stream>>>(y, W, s, b, out, B);
}